// circuit_TNF_2d_Model_44830868636263
// MI455X (gfx1250) — compile-verified
//
#include <hip/hip_runtime.h>

// ---------------------------------------------------------------------------
// PEPS (4x5, D=4, P=2) -> 2^20 statevector via column-wise contraction
// (re-associated to keep intermediates tiny). Heavy contractions are fp32
// WMMA GEMMs whose A/B panels are brought into LDS by the Tensor Data Mover
// (tensor_load_to_lds + s_wait_tensorcnt), with TDM padding used to kill
// LDS bank conflicts. Then 155 in-place two-qubit gate sweeps and a gather.
// ---------------------------------------------------------------------------

typedef float    v2f __attribute__((ext_vector_type(2)));
typedef float    v8f __attribute__((ext_vector_type(8)));
typedef unsigned v4u __attribute__((ext_vector_type(4)));
typedef unsigned v8u __attribute__((ext_vector_type(8)));

#define LXd 4
#define LYd 5
#define NQ  20
#define PSI_N (1u << NQ)

// ---------------------------------------------------------------------------
// Column build step:  out[(P*2), d, (L*l), (R*r)] = sum_u A[P,u,L,R]*site[p,u,d,ls,rs]
// site strides inside one peps site block (2,4,4,4,4): p:256 u:64 d:16 l:4 r:1
// A == nullptr means A is the scalar 1 (first step, all dims 1).
// ---------------------------------------------------------------------------
__global__ void col_step(const float* __restrict__ A, const float* __restrict__ site,
                         float* __restrict__ out,
                         int P, int U, int L, int R,
                         int d, int l, int r, int total) {
    int idx = blockIdx.x * blockDim.x + threadIdx.x;
    if (idx >= total) return;
    int Rr = R * r, Ll = L * l;
    int rr = idx % Rr; int t = idx / Rr;
    int ll = t % Ll;   t /= Ll;
    int dd = t % d;    int pp = t / d;
    int Pprev = pp >> 1, p = pp & 1;
    int Lprev = ll / l, ls = ll % l;
    int Rprev = rr / r, rs = rr % r;
    const float* sb = site + p * 256 + dd * 16 + ls * 4 + rs;
    float acc = 0.f;
    for (int u = 0; u < U; ++u) {
        float av = A ? A[((Pprev * U + u) * L + Lprev) * R + Rprev] : 1.0f;
        acc += av * sb[u * 64];
    }
    out[idx] = acc;
}

// in[P][L][R] -> out[L][P][R]  (L==1 makes this a plain copy)
__global__ void transpose_plr(const float* __restrict__ in, float* __restrict__ out,
                              int P, int L, int R, int total) {
    int idx = blockIdx.x * blockDim.x + threadIdx.x;
    if (idx >= total) return;
    int rr = idx % R; int t = idx / R;
    int ll = t % L;   int pp = t / L;
    out[(ll * P + pp) * R + rr] = in[idx];
}

// ---------------------------------------------------------------------------
// Tensor DMA descriptor (D#) builder: 2D tile of 4-byte elements.
//   group0: count=1 | lds_addr | global_addr | type=2
//   group1: data_size=2 (4B), optional LDS padding, dims/strides in elements.
// tdim0/tdim1 are the in-range extents from the tile origin (OOB reads -> 0).
// ---------------------------------------------------------------------------
__device__ __forceinline__ void tdm_desc2d(v4u& g0, v8u& g1,
                                           unsigned lds_addr,
                                           unsigned long long gaddr,
                                           unsigned tdim0, unsigned tdim1,
                                           unsigned tile0, unsigned tile1,
                                           unsigned long long stride0,
                                           unsigned pad_en, unsigned pad_interval,
                                           unsigned pad_amount) {
    g0[0] = 1u;                                  // count=1, user mode
    g0[1] = lds_addr;                            // lds_addr [63:32]
    g0[2] = (unsigned)(gaddr & 0xffffffffull);   // global_addr [95:64]
    g0[3] = (unsigned)((gaddr >> 32) & 0x1ffffffull) | (2u << 30); // [120:96] | type=2

    g1[0] = (2u << 16) | (pad_en << 20) | (pad_interval << 22) | (pad_amount << 25);
    g1[1] = (tdim0 & 0xffffu) << 16;                                   // td0[15:0]
    g1[2] = ((tdim0 >> 16) & 0xffffu) | ((tdim1 & 0xffffu) << 16);     // td0[31:16]|td1[15:0]
    g1[3] = ((tdim1 >> 16) & 0xffffu) | ((tile0 & 0xffffu) << 16);     // td1[31:16]|tile0
    g1[4] = (tile1 & 0xffffu);                                         // tile1 | tile2=0
    g1[5] = (unsigned)(stride0 & 0xffffffffull);                       // stride0[31:0]
    g1[6] = (unsigned)((stride0 >> 32) & 0xffffull);                   // stride0[47:32]|stride1[15:0]
    g1[7] = 0u;                                                        // stride1[47:16]
}

// ---------------------------------------------------------------------------
// fp32 GEMM, D = A(MxK) * B(KxN), row-major, K == 256, M,N multiples of 16.
// 2 waves/block, each wave owns one 16x16 output tile. Wave 0 issues two TDM
// loads: the full 16x256 A panel (LDS-padded: 260-dword rows => conflict-free
// column reads) and the 256x32 B panel. One s_wait_tensorcnt + one barrier,
// then 64 V_WMMA_F32_16X16X4_F32 straight from LDS, no further barriers.
// ---------------------------------------------------------------------------
#define GEMM_WAVES 2
#define A_STRIDE   260   // 256 + 4 pad dwords per row (TDM pad_interval=7, amount=3)
__global__ __launch_bounds__(64)
void gemm_wmma(const float* __restrict__ A, const float* __restrict__ B,
               float* __restrict__ D, int M, int N, int K) {
    __shared__ float As[16 * A_STRIDE];          // 16.6 KB
    __shared__ float Bs[256 * (16 * GEMM_WAVES)];// 32 KB
    const int tid  = threadIdx.x;
    const int wave = tid >> 5;
    const int lane = tid & 31;
    const int half = lane >> 4;     // 0: lanes 0-15, 1: lanes 16-31
    const int lr   = lane & 15;
    const int m0   = blockIdx.y * 16;
    const int n0b  = blockIdx.x * (16 * GEMM_WAVES);
    const int n0   = n0b + wave * 16;

    if (wave == 0) {
        v4u g0a, g0b; v8u g1a, g1b;
        // A panel: 16 rows x K cols from A[m0][0], row stride K, pad rows to 260 dw
        tdm_desc2d(g0a, g1a,
                   (unsigned)(unsigned long long)(void*)As,
                   (unsigned long long)(const void*)(A + (unsigned long long)m0 * K),
                   (unsigned)K, (unsigned)(M - m0), (unsigned)K, 16u,
                   (unsigned long long)K, 1u, 7u, 3u);
        // B panel: 256 rows x 32 cols from B[0][n0b], row stride N, OOB cols -> 0
        tdm_desc2d(g0b, g1b,
                   (unsigned)(unsigned long long)(void*)Bs,
                   (unsigned long long)(const void*)(B + n0b),
                   (unsigned)(N - n0b), (unsigned)K, 16u * GEMM_WAVES, (unsigned)K,
                   (unsigned long long)N, 0u, 0u, 0u);
        asm volatile("tensor_load_to_lds %0, %1" :: "s"(g0a), "s"(g1a) : "memory");
        asm volatile("tensor_load_to_lds %0, %1" :: "s"(g0b), "s"(g1b) : "memory");
        __builtin_amdgcn_s_wait_tensorcnt(0);
    }
    __syncthreads();

    v8f acc = {};
    const int bcol = wave * 16 + lr;
    #pragma unroll 8
    for (int k0 = 0; k0 < 256; k0 += 4) {
        const int kk = k0 + 2 * half;   // VGPR0 holds K=k0 (lanes 0-15) / K=k0+2 (16-31)
        v2f a, b;
        a.x = As[lr * A_STRIDE + kk];
        a.y = As[lr * A_STRIDE + kk + 1];
        b.x = Bs[kk * (16 * GEMM_WAVES) + bcol];
        b.y = Bs[(kk + 1) * (16 * GEMM_WAVES) + bcol];
        acc = __builtin_amdgcn_wmma_f32_16x16x4_f32(
            false, a, false, b, (short)0, acc, false, false);
    }

    if (n0 < N) {
        #pragma unroll
        for (int g = 0; g < 8; ++g) {
            int row = m0 + g + 8 * half; // C/D layout: VGPR g = M=g / M=g+8
            D[row * N + n0 + lr] = acc[g];
        }
    }
}

// column-major site ordering (bit pos 4*j+i from MSB) -> row-major (5*i+j)
__global__ void permute_rowmajor(const float* __restrict__ pc, float* __restrict__ psi) {
    unsigned t = blockIdx.x * blockDim.x + threadIdx.x;
    unsigned c = 0;
    #pragma unroll
    for (int i = 0; i < LXd; ++i)
        #pragma unroll
        for (int j = 0; j < LYd; ++j) {
            unsigned bit = (t >> (NQ - 1 - (i * LYd + j))) & 1u;
            c |= bit << (NQ - 1 - (j * LXd + i));
        }
    psi[t] = pc[c];
}

// in-place two-qubit gate: plow = 19-b (lower bit pos), phigh = 19-a
__global__ void apply_gate(float* __restrict__ psi, const float* __restrict__ g,
                           int plow, int phigh) {
    unsigned t = blockIdx.x * blockDim.x + threadIdx.x; // < 2^18
    unsigned i1   = ((t  >> plow)  << (plow  + 1)) | (t  & ((1u << plow)  - 1u));
    unsigned base = ((i1 >> phigh) << (phigh + 1)) | (i1 & ((1u << phigh) - 1u));
    unsigned sa = 1u << phigh, sb = 1u << plow;
    float v0 = psi[base];
    float v1 = psi[base + sb];
    float v2 = psi[base + sa];
    float v3 = psi[base + sa + sb];
    float g00=g[0],g01=g[1],g02=g[2],g03=g[3],
          g10=g[4],g11=g[5],g12=g[6],g13=g[7],
          g20=g[8],g21=g[9],g22=g[10],g23=g[11],
          g30=g[12],g31=g[13],g32=g[14],g33=g[15];
    psi[base]           = g00*v0 + g01*v1 + g02*v2 + g03*v3;
    psi[base + sb]      = g10*v0 + g11*v1 + g12*v2 + g13*v3;
    psi[base + sa]      = g20*v0 + g21*v1 + g22*v2 + g23*v3;
    psi[base + sa + sb] = g30*v0 + g31*v1 + g32*v2 + g33*v3;
}

__global__ void gather_out(const int* __restrict__ x, const float* __restrict__ psi,
                           float* __restrict__ out, int Bn) {
    int b = blockIdx.x * blockDim.x + threadIdx.x;
    if (b >= Bn) return;
    unsigned idx = 0;
    #pragma unroll
    for (int k = 0; k < NQ; ++k)
        idx |= ((unsigned)x[b * NQ + k] & 1u) << (NQ - 1 - k);
    out[b] = psi[idx];
}

extern "C" void kernel_launch(void* const* d_in, const int* in_sizes, int n_in,
                              void* d_out, int out_size, void* d_ws, size_t ws_size,
                              hipStream_t stream) {
    const int*   x    = (const int*)d_in[0];     // (64, 20) int32
    const float* peps = (const float*)d_in[1];   // (4,5,2,4,4,4,4) f32
    const float* gate = (const float*)d_in[2];   // (4,4) f32
    float* out = (float*)d_out;
    float* ws  = (float*)d_ws;

    const unsigned MEG = 1u << 20;
    float* bufA = ws;
    float* bufB = ws + MEG;
    float* C0   = ws + 2 * MEG;          // (16,256)        A-layout
    float* C1   = C0 + 4096;             // (256,16,256)    [L][P][R]
    float* C2   = C1 + MEG;
    float* C3   = C2 + MEG;
    float* C4   = C3 + MEG;              // (256,16)        [L][P]
    float* S1   = C4 + 4096;             // 16 x 4096  == 256 x 256
    float* Wb   = S1 + 65536;            // 4096 x 16  == 256 x 256
    float* PSI  = Wb + 65536;            // 2^20 row-major statevector
    float* S2   = bufA;                  // reuse: 256 x 4096 == 4096 x 256
    float* PC   = bufB;                  // reuse: psi, column-major ordering

    // ---- build the 5 column tensors -------------------------------------
    for (int j = 0; j < LYd; ++j) {
        int l = (j == 0) ? 1 : 4;
        int r = (j == LYd - 1) ? 1 : 4;
        int P = 1, L = 1, R = 1;
        const float* Aptr = nullptr;
        float* bufs[2] = {bufA, bufB};
        int pp = 0;
        for (int i = 0; i < LXd; ++i) {
            int u = (i == 0) ? 1 : 4;
            int d = (i == LXd - 1) ? 1 : 4;
            const float* site = peps + (i * LYd + j) * 512;
            int total = (P * 2) * d * (L * l) * (R * r);
            float* obuf = bufs[pp];
            col_step<<<(total + 255) / 256, 256, 0, stream>>>(
                Aptr, site, obuf, P, u, L, R, d, l, r, total);
            P *= 2; L *= l; R *= r;
            Aptr = obuf; pp ^= 1;
        }
        float* dst = (j == 0) ? C0 : (j == 1) ? C1 : (j == 2) ? C2 : (j == 3) ? C3 : C4;
        int tot = 16 * L * R;
        transpose_plr<<<(tot + 255) / 256, 256, 0, stream>>>(Aptr, dst, 16, L, R, tot);
    }

    // ---- contraction as 4 fp32 WMMA GEMMs (K = 256 everywhere) ----------
    // S1(16x4096)  = C0(16x256) * C1(256x4096)
    gemm_wmma<<<dim3(128, 1),  64, 0, stream>>>(C0, C1, S1, 16,   4096, 256);
    // S2(256x4096) = S1(256x256) * C2(256x4096)
    gemm_wmma<<<dim3(128, 16), 64, 0, stream>>>(S1, C2, S2, 256,  4096, 256);
    // W(4096x16)   = C3(4096x256) * C4(256x16)
    gemm_wmma<<<dim3(1, 256),  64, 0, stream>>>(C3, C4, Wb, 4096, 16,   256);
    // psi_col(4096x256) = S2(4096x256) * W(256x256)
    gemm_wmma<<<dim3(8, 256),  64, 0, stream>>>(S2, Wb, PC, 4096, 256,  256);

    // ---- reorder bits: column-major sites -> row-major sites ------------
    permute_rowmajor<<<PSI_N / 256, 256, 0, stream>>>(PC, PSI);

    // ---- 5 sweeps of 31 two-qubit gates, in place -----------------------
    for (int dep = 0; dep < 5; ++dep) {
        for (int i = 0; i < LXd; ++i)
            for (int j = 0; j < LYd - 1; ++j) {
                int a = i * LYd + j, b = a + 1;
                apply_gate<<<1024, 256, 0, stream>>>(PSI, gate, NQ - 1 - b, NQ - 1 - a);
            }
        for (int i = 0; i < LXd - 1; ++i)
            for (int j = 0; j < LYd; ++j) {
                int a = i * LYd + j, b = a + LYd;
                apply_gate<<<1024, 256, 0, stream>>>(PSI, gate, NQ - 1 - b, NQ - 1 - a);
            }
    }

    // ---- gather the 64 requested amplitudes -----------------------------
    gather_out<<<1, 64, 0, stream>>>(x, PSI, out, out_size);
}